// mmWaveSimulator_45165876085271
// MI455X (gfx1250) — compile-verified
//
#include <hip/hip_runtime.h>
#include <hip/hip_bf16.h>
#include <math.h>

#define PI_F 3.14159265358979323846f
#define C_LIGHT 3.0e8f
#define TWO_PI_OVER_C (2.0f * PI_F / C_LIGHT)

typedef __attribute__((ext_vector_type(2))) float v2f;
typedef __attribute__((ext_vector_type(8))) float v8f;

// ---- WMMA helper: D = A(16x4 f32) * B(4x16 f32) + C (16x16 f32) ----
__device__ __forceinline__ v8f wmma4(v2f a, v2f b, v8f c) {
    return __builtin_amdgcn_wmma_f32_16x16x4_f32(
        /*neg_a=*/false, a, /*neg_b=*/false, b,
        /*c_mod=*/(short)0, c, /*reuse_a=*/false, /*reuse_b=*/false);
}

// ---------------------------------------------------------------------------
// K1: phase[m] = exp(i * clip(meta_theta[m], -pi, 20pi/180))
__global__ void kPhase(const float* __restrict__ mt,
                       float* __restrict__ phc, float* __restrict__ phs) {
    int m = blockIdx.x * blockDim.x + threadIdx.x;
    if (m >= 3600) return;
    float th = fminf(fmaxf(mt[m], -PI_F), 20.0f * PI_F / 180.0f);
    float sn, cs;
    __sincosf(th, &sn, &cs);
    phc[m] = cs;
    phs[m] = sn;
}

// ---------------------------------------------------------------------------
// K2: B3[f][m][r] = phase[m] * H4[f,m,r],  H4 = channel(meta, rx)
__global__ void kB3(const float* __restrict__ fArr, const float* __restrict__ rx,
                    const float* __restrict__ meta,
                    const float* __restrict__ phc, const float* __restrict__ phs,
                    float* __restrict__ B3re, float* __restrict__ B3im) {
    int tid = blockIdx.x * blockDim.x + threadIdx.x;
    if (tid >= 5 * 3600 * 16) return;
    int r = tid & 15;
    int m = (tid >> 4) % 3600;
    int f = tid / (3600 * 16);
    float dx = meta[m * 3 + 0] - rx[r * 3 + 0];
    float dy = meta[m * 3 + 1] - rx[r * 3 + 1];
    float dz = meta[m * 3 + 2] - rx[r * 3 + 2];
    float d2 = dx * dx + dy * dy + dz * dz;
    float d = sqrtf(d2);
    float amp = 0.001f / d2;
    float phi = TWO_PI_OVER_C * fArr[f] * d;
    float sn, cs;
    __sincosf(phi, &sn, &cs);
    float hr = amp * cs, hi = -amp * sn;   // amp * exp(-i phi)
    float pc = phc[m], ps = phs[m];
    B3re[tid] = pc * hr - ps * hi;
    B3im[tid] = pc * hi + ps * hr;
}

// ---------------------------------------------------------------------------
// K3: H3[f][s][m] = channel(scene, meta)  (== H2[f][m][s] by distance symmetry)
__global__ void kH3(const float* __restrict__ fArr, const float* __restrict__ scene,
                    const float* __restrict__ meta,
                    float* __restrict__ H3re, float* __restrict__ H3im) {
    int tid = blockIdx.x * blockDim.x + threadIdx.x;
    if (tid >= 5 * 400 * 3600) return;
    int m = tid % 3600;
    int s = (tid / 3600) % 400;
    int f = tid / 1440000;
    float dx = scene[s * 3 + 0] - meta[m * 3 + 0];
    float dy = scene[s * 3 + 1] - meta[m * 3 + 1];
    float dz = scene[s * 3 + 2] - meta[m * 3 + 2];
    float d2 = dx * dx + dy * dy + dz * dz;
    float d = sqrtf(d2);
    float amp = 0.001f / d2;
    float phi = TWO_PI_OVER_C * fArr[f] * d;
    float sn, cs;
    __sincosf(phi, &sn, &cs);
    H3re[tid] = amp * cs;
    H3im[tid] = -amp * sn;
}

// ---------------------------------------------------------------------------
// K4: A1[f][k][m] = (sum_t exp(i w[f,k,t]) * H1[f,t,m]) * phase[m]
__global__ void kA1(const float* __restrict__ fArr, const float* __restrict__ w,
                    const float* __restrict__ tx, const float* __restrict__ meta,
                    const float* __restrict__ phc, const float* __restrict__ phs,
                    float* __restrict__ A1re, float* __restrict__ A1im) {
    int tid = blockIdx.x * blockDim.x + threadIdx.x;
    if (tid >= 5 * 3600) return;
    int m = tid % 3600;
    int f = tid / 3600;
    float fv = fArr[f];
    float mx = meta[m * 3 + 0], my = meta[m * 3 + 1], mz = meta[m * 3 + 2];
    float h1r[12], h1i[12];
#pragma unroll
    for (int t = 0; t < 12; ++t) {
        float dx = tx[t * 3 + 0] - mx;
        float dy = tx[t * 3 + 1] - my;
        float dz = tx[t * 3 + 2] - mz;
        float d2 = dx * dx + dy * dy + dz * dz;
        float d = sqrtf(d2);
        float amp = 0.001f / d2;
        float phi = TWO_PI_OVER_C * fv * d;
        float sn, cs;
        __sincosf(phi, &sn, &cs);
        h1r[t] = amp * cs;
        h1i[t] = -amp * sn;
    }
    float pc = phc[m], ps = phs[m];
#pragma unroll
    for (int k = 0; k < 6; ++k) {
        float ar = 0.f, ai = 0.f;
#pragma unroll
        for (int t = 0; t < 12; ++t) {
            float wv = w[f * 72 + k * 12 + t];
            float sw, cw;
            __sincosf(wv, &sw, &cw);          // pri = cw + i sw
            ar += cw * h1r[t] - sw * h1i[t];
            ai += cw * h1i[t] + sw * h1r[t];
        }
        A1re[f * 21600 + k * 3600 + m] = ar * pc - ai * ps;
        A1im[f * 21600 + k * 3600 + m] = ar * ps + ai * pc;
    }
}

// ---------------------------------------------------------------------------
// K5: G[f][s][r] = sum_m H3[f,s,m] * B3[f,m,r]   (complex; 4 real WMMA chains)
// grid: 125 blocks (f=blk/25, s_tile=blk%25), 128 threads = 4 waves, K split 4x.
__global__ __launch_bounds__(128) void kG(const float* __restrict__ H3re,
                                          const float* __restrict__ H3im,
                                          const float* __restrict__ B3re,
                                          const float* __restrict__ B3im,
                                          float* __restrict__ Gre,
                                          float* __restrict__ Gim) {
    const int f    = blockIdx.x / 25;
    const int st   = blockIdx.x % 25;
    const int lane = threadIdx.x & 31;
    const int wave = threadIdx.x >> 5;
    const int srow = st * 16 + (lane & 15);   // A fragment: M = scene row
    const int rcol = lane & 15;               // B fragment: N = rx col
    const int koff = (lane >> 4) << 1;        // 0 or 2
    const float* Ar = H3re + (size_t)f * 1440000 + (size_t)srow * 3600;
    const float* Ai = H3im + (size_t)f * 1440000 + (size_t)srow * 3600;
    const float* Br = B3re + f * 57600;
    const float* Bi = B3im + f * 57600;
    v8f acc[4] = {v8f{}, v8f{}, v8f{}, v8f{}};
    const int kb = wave * 900;
    for (int k0 = kb; k0 < kb + 900; k0 += 4) {
        int ka = k0 + koff;
        v2f ar = *(const v2f*)(Ar + ka);
        v2f ai = *(const v2f*)(Ai + ka);
        v2f br, bi;
        br.x = Br[ka * 16 + rcol];
        br.y = Br[(ka + 1) * 16 + rcol];
        bi.x = Bi[ka * 16 + rcol];
        bi.y = Bi[(ka + 1) * 16 + rcol];
        acc[0] = wmma4(ar, br, acc[0]);   // Re*Re
        acc[1] = wmma4(ai, bi, acc[1]);   // Im*Im
        acc[2] = wmma4(ar, bi, acc[2]);   // Re*Im
        acc[3] = wmma4(ai, br, acc[3]);   // Im*Re
    }
    __shared__ float lds[4 * 4 * 256];    // [wave][acc][elem]
#pragma unroll
    for (int q = 0; q < 4; ++q)
#pragma unroll
        for (int v = 0; v < 8; ++v)
            lds[(wave * 4 + q) * 256 + v * 32 + lane] = acc[q][v];
    __syncthreads();
#pragma unroll
    for (int i = 0; i < 2; ++i) {
        int e = threadIdx.x + i * 128;
        float s0 = 0.f, s1 = 0.f, s2 = 0.f, s3 = 0.f;
#pragma unroll
        for (int ww = 0; ww < 4; ++ww) {
            s0 += lds[(ww * 4 + 0) * 256 + e];
            s1 += lds[(ww * 4 + 1) * 256 + e];
            s2 += lds[(ww * 4 + 2) * 256 + e];
            s3 += lds[(ww * 4 + 3) * 256 + e];
        }
        int v = e >> 5, l = e & 31;
        int s = st * 16 + v + ((l & 16) ? 8 : 0);
        int r = l & 15;
        Gre[f * 6400 + s * 16 + r] = s0 - s1;
        Gim[f * 6400 + s * 16 + r] = s2 + s3;
    }
}

// ---------------------------------------------------------------------------
// K6: A2[f][k][s] = sum_m A1[f,k,m] * H3[f,s,m]   (A1 rows padded 6->16)
__global__ __launch_bounds__(128) void kA2(const float* __restrict__ A1re,
                                           const float* __restrict__ A1im,
                                           const float* __restrict__ H3re,
                                           const float* __restrict__ H3im,
                                           float* __restrict__ A2re,
                                           float* __restrict__ A2im) {
    const int f    = blockIdx.x / 25;
    const int st   = blockIdx.x % 25;
    const int lane = threadIdx.x & 31;
    const int wave = threadIdx.x >> 5;
    const int krow = lane & 15;               // A fragment: M = code row (valid < 6)
    const int scol = st * 16 + (lane & 15);   // B fragment: N = scene col
    const int koff = (lane >> 4) << 1;
    const bool rowOk = krow < 6;
    const float* Ar = A1re + f * 21600 + krow * 3600;
    const float* Ai = A1im + f * 21600 + krow * 3600;
    const float* Br = H3re + (size_t)f * 1440000 + (size_t)scol * 3600;  // B[k][n]=H3[n][k]
    const float* Bi = H3im + (size_t)f * 1440000 + (size_t)scol * 3600;
    v8f acc[4] = {v8f{}, v8f{}, v8f{}, v8f{}};
    const int kb = wave * 900;
    for (int k0 = kb; k0 < kb + 900; k0 += 4) {
        int ka = k0 + koff;
        v2f ar = {0.f, 0.f}, ai = {0.f, 0.f};
        if (rowOk) {
            ar = *(const v2f*)(Ar + ka);
            ai = *(const v2f*)(Ai + ka);
        }
        v2f br = *(const v2f*)(Br + ka);
        v2f bi = *(const v2f*)(Bi + ka);
        acc[0] = wmma4(ar, br, acc[0]);
        acc[1] = wmma4(ai, bi, acc[1]);
        acc[2] = wmma4(ar, bi, acc[2]);
        acc[3] = wmma4(ai, br, acc[3]);
    }
    __shared__ float lds[4 * 4 * 256];
#pragma unroll
    for (int q = 0; q < 4; ++q)
#pragma unroll
        for (int v = 0; v < 8; ++v)
            lds[(wave * 4 + q) * 256 + v * 32 + lane] = acc[q][v];
    __syncthreads();
#pragma unroll
    for (int i = 0; i < 2; ++i) {
        int e = threadIdx.x + i * 128;
        float s0 = 0.f, s1 = 0.f, s2 = 0.f, s3 = 0.f;
#pragma unroll
        for (int ww = 0; ww < 4; ++ww) {
            s0 += lds[(ww * 4 + 0) * 256 + e];
            s1 += lds[(ww * 4 + 1) * 256 + e];
            s2 += lds[(ww * 4 + 2) * 256 + e];
            s3 += lds[(ww * 4 + 3) * 256 + e];
        }
        int v = e >> 5, l = e & 31;
        int k = v + ((l & 16) ? 8 : 0);
        if (k < 6) {
            int s = st * 16 + (l & 15);
            A2re[f * 2400 + k * 400 + s] = s0 - s1;
            A2im[f * 2400 + k * 400 + s] = s2 + s3;
        }
    }
}

// ---------------------------------------------------------------------------
// K7: AF[u][s] = A2[f,k,s]*G[f,s,r] (u = r*30+f*6+k); write s_A with 12x tile.
__global__ void kOut1(const float* __restrict__ A2re, const float* __restrict__ A2im,
                      const float* __restrict__ Gre, const float* __restrict__ Gim,
                      float* __restrict__ sA,
                      float* __restrict__ AFre, float* __restrict__ AFim) {
    int tid = blockIdx.x * blockDim.x + threadIdx.x;
    if (tid >= 5 * 6 * 16 * 400) return;
    int s  = tid % 400;
    int t2 = tid / 400;
    int r  = t2 & 15;
    int t3 = t2 >> 4;       // f*6 + k
    int k  = t3 % 6;
    int f  = t3 / 6;
    float a2r = A2re[f * 2400 + k * 400 + s];
    float a2i = A2im[f * 2400 + k * 400 + s];
    float gr  = Gre[f * 6400 + s * 16 + r];
    float gi  = Gim[f * 6400 + s * 16 + r];
    float cr = a2r * gr - a2i * gi;
    float ci = a2r * gi + a2i * gr;
    int u = r * 30 + f * 6 + k;
    AFre[u * 400 + s] = cr;
    AFim[u * 400 + s] = ci;
    int base = r * 360 + f * 72 + k;
#pragma unroll
    for (int rep = 0; rep < 12; ++rep) {
        int row = base + rep * 6;
        sA[row * 400 + s]            = cr;
        sA[(row + 5760) * 400 + s]   = ci;
    }
}

// ---------------------------------------------------------------------------
// K8: mic[u][b] = sum_s AF[u][s] * T[s][b]  (real/imag WMMA), replicated 12x.
// grid: 30 blocks x 256 threads = 8 waves; tile = blk*8+wave (30 u-tiles x 8 b-tiles)
__global__ __launch_bounds__(256) void kMic(const float* __restrict__ AFre,
                                            const float* __restrict__ AFim,
                                            const float* __restrict__ T,
                                            float* __restrict__ outMic) {
    const int wave = threadIdx.x >> 5;
    const int lane = threadIdx.x & 31;
    const int tile = blockIdx.x * 8 + wave;   // 0..239
    const int ut   = tile >> 3;               // 0..29
    const int bt   = tile & 7;                // 0..7
    const int urow = ut * 16 + (lane & 15);
    const int bcol = bt * 16 + (lane & 15);
    const int koff = (lane >> 4) << 1;
    const float* Ar = AFre + urow * 400;
    const float* Ai = AFim + urow * 400;
    v8f accR = {}, accI = {};
    for (int k0 = 0; k0 < 400; k0 += 4) {
        int ka = k0 + koff;
        v2f ar = *(const v2f*)(Ar + ka);
        v2f ai = *(const v2f*)(Ai + ka);
        v2f b;
        b.x = T[ka * 128 + bcol];
        b.y = T[(ka + 1) * 128 + bcol];
        accR = wmma4(ar, b, accR);
        accI = wmma4(ai, b, accI);
    }
#pragma unroll
    for (int v = 0; v < 8; ++v) {
        int u = ut * 16 + v + ((lane & 16) ? 8 : 0);
        int rx = u / 30, rem = u % 30;
        int f = rem / 6, k = rem % 6;
        int base = rx * 360 + f * 72 + k;
        float vr = accR[v], vi = accI[v];
#pragma unroll
        for (int rep = 0; rep < 12; ++rep) {
            int row = base + rep * 6;
            outMic[row * 128 + bcol]            = vr;
            outMic[(row + 5760) * 128 + bcol]   = vi;
        }
    }
}

// ---------------------------------------------------------------------------
extern "C" void kernel_launch(void* const* d_in, const int* in_sizes, int n_in,
                              void* d_out, int out_size, void* d_ws, size_t ws_size,
                              hipStream_t stream) {
    const float* T     = (const float*)d_in[0];   // [400,128]
    const float* w     = (const float*)d_in[1];   // [5,6,12]
    const float* mt    = (const float*)d_in[2];   // [3600,1]
    const float* fArr  = (const float*)d_in[3];   // [5]
    const float* tx    = (const float*)d_in[4];   // [12,3]
    const float* rx    = (const float*)d_in[5];   // [16,3]
    const float* meta  = (const float*)d_in[6];   // [3600,3]
    const float* scene = (const float*)d_in[7];   // [400,3]
    float* out = (float*)d_out;
    float* sA  = out;                  // [11520,400]
    float* mic = out + 4608000;        // [11520,128]

    // workspace layout (floats), total ~15.7M floats (~63 MB, L2-resident)
    float* ws   = (float*)d_ws;
    float* phc  = ws;                  // 3600
    float* phs  = phc  + 3600;         // 3600
    float* H3re = phs  + 3600;         // 7,200,000
    float* H3im = H3re + 7200000;      // 7,200,000
    float* B3re = H3im + 7200000;      // 288,000
    float* B3im = B3re + 288000;       // 288,000
    float* A1re = B3im + 288000;       // 108,000
    float* A1im = A1re + 108000;       // 108,000
    float* Gre  = A1im + 108000;       // 32,000
    float* Gim  = Gre  + 32000;        // 32,000
    float* A2re = Gim  + 32000;        // 12,000
    float* A2im = A2re + 12000;        // 12,000
    float* AFre = A2im + 12000;        // 192,000
    float* AFim = AFre + 192000;       // 192,000

    kPhase<<<15, 256, 0, stream>>>(mt, phc, phs);
    kB3<<<1125, 256, 0, stream>>>(fArr, rx, meta, phc, phs, B3re, B3im);
    kH3<<<28125, 256, 0, stream>>>(fArr, scene, meta, H3re, H3im);
    kA1<<<71, 256, 0, stream>>>(fArr, w, tx, meta, phc, phs, A1re, A1im);
    kG<<<125, 128, 0, stream>>>(H3re, H3im, B3re, B3im, Gre, Gim);
    kA2<<<125, 128, 0, stream>>>(A1re, A1im, H3re, H3im, A2re, A2im);
    kOut1<<<750, 256, 0, stream>>>(A2re, A2im, Gre, Gim, sA, AFre, AFim);
    kMic<<<30, 256, 0, stream>>>(AFre, AFim, T, mic);
}